// OrganADR_25778393710991
// MI455X (gfx1250) — compile-verified
//
#include <hip/hip_runtime.h>
#include <hip/hip_bf16.h>

typedef __attribute__((ext_vector_type(2))) float v2f;
typedef __attribute__((ext_vector_type(8))) float v8f;

#define N_ENT 2000
#define N_REL 386
#define N_DIM 64
#define NLAY  3
#define BSZ   64
#define NEDGE 20000
#define ROW   4096   // B*D per entity row

// ---------------------------------------------------------------------------
// 1) Entity embedding GEMM: (128 rows, 1024) x (1024, 64) -> ht (64, 128)
//    One wave per 16x16 tile; fp32 WMMA 16x16x4, 256 chained k-steps.
// ---------------------------------------------------------------------------
__global__ void embed_kernel(const int* __restrict__ head, const int* __restrict__ tail,
                             const float* __restrict__ entkg,
                             const float* __restrict__ Ww, const float* __restrict__ Wb,
                             float* __restrict__ ht) {
  int mtile = blockIdx.x;          // 0..7  (rows 0..63 = head, 64..127 = tail)
  int ntile = blockIdx.y;          // 0..3
  int lane  = threadIdx.x;         // 32 threads = 1 wave, EXEC all ones
  int mrow  = mtile * 16 + (lane & 15);
  int g     = lane >> 4;           // lane half selects K pair
  int n     = ntile * 16 + (lane & 15);
  int ent   = (mrow < BSZ) ? head[mrow] : tail[mrow - BSZ];
  const float* arow = entkg + (size_t)ent * 1024;

  v8f c = {};
  for (int k0 = 0; k0 < 1024; k0 += 4) {
    v2f a = *(const v2f*)(arow + k0 + 2 * g);          // A[m][k], A[m][k+1]
    v2f b;
    b.x = Ww[(size_t)(k0 + 2 * g + 0) * N_DIM + n];
    b.y = Ww[(size_t)(k0 + 2 * g + 1) * N_DIM + n];
    c = __builtin_amdgcn_wmma_f32_16x16x4_f32(false, a, false, b, (short)0, c, false, false);
  }
  float bn = Wb[n];
#pragma unroll
  for (int r = 0; r < 8; ++r) {
    int row = mtile * 16 + r + 8 * g;
    float v = c[r] + bn;
    if (row < BSZ) ht[(size_t)row * 128 + n] = v;                       // head half
    else           ht[(size_t)(row - BSZ) * 128 + N_DIM + n] = v;      // tail half
  }
}

// ---------------------------------------------------------------------------
// 2) Relation attention, written TRANSPOSED: rwT[l][r][b]
//    rw[l,b,r] = sigmoid(relu(ht[b] @ rl_w[l] + rl_b) @ ar_w[l] + ar_b)
// ---------------------------------------------------------------------------
__global__ void rw_kernel(const float* __restrict__ ht,
                          const float* __restrict__ rl_w, const float* __restrict__ rl_b,
                          const float* __restrict__ ar_w, const float* __restrict__ ar_b,
                          float* __restrict__ rwT) {
  __shared__ float red[128];
  __shared__ float t[5];
  int l = blockIdx.x >> 6;
  int b = blockIdx.x & 63;
  int tid = threadIdx.x;
  float x = ht[(size_t)b * 128 + tid];
  const float* rlw = rl_w + (size_t)l * 128 * 5;
  for (int j = 0; j < 5; ++j) {
    red[tid] = x * rlw[tid * 5 + j];
    __syncthreads();
    for (int s = 64; s > 0; s >>= 1) {
      if (tid < s) red[tid] += red[tid + s];
      __syncthreads();
    }
    if (tid == 0) { float v = red[0] + rl_b[l * 5 + j]; t[j] = v > 0.f ? v : 0.f; }
    __syncthreads();
  }
  for (int r = tid; r < N_REL; r += 128) {
    float s = ar_b[(size_t)l * N_REL + r];
#pragma unroll
    for (int j = 0; j < 5; ++j) s += t[j] * ar_w[((size_t)l * 5 + j) * N_REL + r];
    rwT[((size_t)l * N_REL + r) * BSZ + b] = 1.f / (1.f + __expf(-s));
  }
}

// ---------------------------------------------------------------------------
// 3) CSR build (deterministic: within-row order == edge-index order)
// ---------------------------------------------------------------------------
__global__ void zero_int_kernel(int* p, int n) {
  int i = blockIdx.x * blockDim.x + threadIdx.x;
  if (i < n) p[i] = 0;
}
__global__ void csr_count_kernel(const int* __restrict__ dst, int* __restrict__ counts, int E) {
  int e = blockIdx.x * blockDim.x + threadIdx.x;
  if (e < E) atomicAdd(&counts[dst[e]], 1);
}
__global__ void csr_scan_kernel(const int* __restrict__ counts, int* __restrict__ rowst, int n) {
  if (threadIdx.x == 0 && blockIdx.x == 0) {
    int acc = 0;
    for (int i = 0; i < n; ++i) { rowst[i] = acc; acc += counts[i]; }
    rowst[n] = acc;
  }
}
__global__ void csr_fill_kernel(const int* __restrict__ dst, const int* __restrict__ rowst,
                                int* __restrict__ csr, int nent, int E) {
  int i = blockIdx.x * blockDim.x + threadIdx.x;
  if (i >= nent) return;
  int p = rowst[i];
  for (int e = 0; e < E; ++e)
    if (dst[e] == i) csr[p++] = e;
}

// ---------------------------------------------------------------------------
// 4) hid init: zero + scatter init rows
// ---------------------------------------------------------------------------
__global__ void zero_f4_kernel(float4* __restrict__ p) {
  size_t i = (size_t)blockIdx.x * blockDim.x + threadIdx.x;
  float4 z; z.x = z.y = z.z = z.w = 0.f;
  p[i] = z;
}
__global__ void scatter_init_kernel(float* __restrict__ hid, const float* __restrict__ ht,
                                    const int* __restrict__ init_idx, int init_off) {
  int idx = blockIdx.x * blockDim.x + threadIdx.x;  // 4096 = B*D
  if (idx >= ROW) return;
  int b = idx >> 6, d = idx & 63;
  hid[(size_t)init_idx[b] * ROW + idx] = ht[(size_t)b * 128 + init_off + d];
}

// ---------------------------------------------------------------------------
// 5) Edge aggregation (bandwidth-dominant; exploits rank-1 structure of
//    rel_in[r,b,d] = rwT[r][b] * rel_kg[r][d] so each edge reads the 16KB
//    hid row plus only 2x256B relation rows instead of a 16KB product row).
//    One block per destination entity; 256 threads x 4 float4 = 4096 lanes.
//    The next edge's hid row is software-prefetched (global_prefetch_b8)
//    at 64B/lane granularity to hide the L2 gather latency.
// ---------------------------------------------------------------------------
__global__ void aggregate_kernel(const float* __restrict__ hid,
                                 const float* __restrict__ relkg,  // layer slice (386, 64)
                                 const float* __restrict__ rwT,    // layer slice (386, 64)
                                 const float* __restrict__ edge_w,
                                 const int* __restrict__ edge_src,
                                 const int* __restrict__ edge_rel,
                                 const int* __restrict__ rowst,
                                 const int* __restrict__ csr,
                                 float* __restrict__ out) {
  int i = blockIdx.x;
  int t = threadIdx.x;
  float4 acc[4];
#pragma unroll
  for (int k = 0; k < 4; ++k) { acc[k].x = acc[k].y = acc[k].z = acc[k].w = 0.f; }
  int beg = rowst[i], end = rowst[i + 1];
  for (int p = beg; p < end; ++p) {
    int e = csr[p];
    if (p + 1 < end) {   // prefetch next row: 256 threads x 64B = full 16KB
      int e2 = csr[p + 1];
      __builtin_prefetch(hid + (size_t)edge_src[e2] * ROW + t * 16, 0, 3);
    }
    float w = edge_w[e];
    int r = edge_rel[e];
    const float4* hp = (const float4*)(hid + (size_t)edge_src[e] * ROW);
    const float4* rk = (const float4*)(relkg + (size_t)r * N_DIM);
    const float*  rb = rwT + (size_t)r * BSZ;
#pragma unroll
    for (int k = 0; k < 4; ++k) {
      int f = k * 256 + t;              // float4 index in the 4096-wide row
      float4 h = hp[f];
      float4 g = rk[f & 15];            // d/4  (a float4 never crosses a b boundary)
      float  s = w * rb[f >> 4];        // b
      acc[k].x += s * g.x * h.x;
      acc[k].y += s * g.y * h.y;
      acc[k].z += s * g.z * h.z;
      acc[k].w += s * g.w * h.w;
    }
  }
  float4* op = (float4*)(out + (size_t)i * ROW);
#pragma unroll
  for (int k = 0; k < 4; ++k) op[k * 256 + t] = acc[k];
}

// ---------------------------------------------------------------------------
// 6) Per-layer linear: Out = relu(A(128000,64) @ W(64,64) + bias), fp32 WMMA.
//    A tile (16x64, contiguous 4KB) staged once per block into LDS with
//    CDNA5 async global->LDS b128 copies (ASYNCcnt path, no VGPR round-trip);
//    stride-68 padding makes the fragment ds-reads bank-conflict-free.
//    Block = 4 waves; wave w owns N-tile w; 16 chained 16x16x4 WMMAs.
// ---------------------------------------------------------------------------
#define ASTRIDE 68
__global__ void lin_gemm_kernel(const float* __restrict__ A, const float* __restrict__ W,
                                const float* __restrict__ bias, float* __restrict__ Out) {
  __shared__ float As[16 * ASTRIDE];
  int mtile = blockIdx.x;            // 0..7999
  int wave  = threadIdx.x >> 5;      // 0..3 -> n-tile
  int lane  = threadIdx.x & 31;
  int g = lane >> 4;
  int n = wave * 16 + (lane & 15);

  // async-stage the contiguous 16x64 A tile: 128 threads x 2 b128 transfers
  const float* gsrc = A + (size_t)mtile * 16 * N_DIM;
#pragma unroll
  for (int q = threadIdx.x; q < 256; q += 128) {
    int row = q >> 4;          // (q*4)/64
    int col = (q & 15) * 4;
    unsigned lds_off = (unsigned)(size_t)&As[row * ASTRIDE + col];
    const float* ga = gsrc + q * 4;
    asm volatile("global_load_async_to_lds_b128 %0, %1, off"
                 :: "v"(lds_off), "v"(ga) : "memory");
  }
  asm volatile("s_wait_asynccnt 0x0" ::: "memory");
  __syncthreads();

  v8f c = {};
#pragma unroll
  for (int k0 = 0; k0 < N_DIM; k0 += 4) {
    v2f a = *(const v2f*)(&As[(lane & 15) * ASTRIDE + k0 + 2 * g]);
    v2f b;
    b.x = W[(k0 + 2 * g + 0) * N_DIM + n];
    b.y = W[(k0 + 2 * g + 1) * N_DIM + n];
    c = __builtin_amdgcn_wmma_f32_16x16x4_f32(false, a, false, b, (short)0, c, false, false);
  }
  float bn = bias[n];
#pragma unroll
  for (int r = 0; r < 8; ++r) {
    int row = mtile * 16 + r + 8 * g;
    float v = c[r] + bn;
    Out[(size_t)row * N_DIM + n] = v > 0.f ? v : 0.f;
  }
}

// ---------------------------------------------------------------------------
// 7) Gather result rows into emb(B,128)
// ---------------------------------------------------------------------------
__global__ void gather_kernel(const float* __restrict__ hid, const int* __restrict__ gidx,
                              float* __restrict__ emb, int emb_off) {
  int idx = blockIdx.x * blockDim.x + threadIdx.x;  // 4096
  if (idx >= ROW) return;
  int b = idx >> 6, d = idx & 63;
  emb[(size_t)b * 128 + emb_off + d] = hid[(size_t)gidx[b] * ROW + idx];
}

// ---------------------------------------------------------------------------
// 8) Final scores: (64,128) @ (128,15) + b
// ---------------------------------------------------------------------------
__global__ void final_kernel(const float* __restrict__ emb, const float* __restrict__ Ww,
                             const float* __restrict__ Wb, float* __restrict__ out) {
  int idx = blockIdx.x * blockDim.x + threadIdx.x;
  if (idx >= BSZ * 15) return;
  int b = idx / 15, j = idx % 15;
  float s = Wb[j];
  for (int k = 0; k < 128; ++k) s += emb[(size_t)b * 128 + k] * Ww[k * 15 + j];
  out[idx] = s;
}

// ---------------------------------------------------------------------------
extern "C" void kernel_launch(void* const* d_in, const int* in_sizes, int n_in,
                              void* d_out, int out_size, void* d_ws, size_t ws_size,
                              hipStream_t stream) {
  (void)in_sizes; (void)n_in; (void)out_size; (void)ws_size;
  const int*   head     = (const int*)d_in[0];
  const int*   tail     = (const int*)d_in[1];
  const int*   edge_dst = (const int*)d_in[2];
  const int*   edge_src = (const int*)d_in[3];
  const int*   edge_rel = (const int*)d_in[4];
  const float* edge_w   = (const float*)d_in[5];
  const float* ent_kg   = (const float*)d_in[6];
  const float* Went_w   = (const float*)d_in[7];
  const float* Went_b   = (const float*)d_in[8];
  const float* rel_kg   = (const float*)d_in[9];
  const float* lin_w    = (const float*)d_in[10];
  const float* lin_b    = (const float*)d_in[11];
  const float* rl_w     = (const float*)d_in[12];
  const float* rl_b     = (const float*)d_in[13];
  const float* ar_w     = (const float*)d_in[14];
  const float* ar_b     = (const float*)d_in[15];
  const float* Wr1_w    = (const float*)d_in[16];
  const float* Wr1_b    = (const float*)d_in[17];
  float* out = (float*)d_out;

  // workspace carve-up (all chunks 16B-aligned)
  char* ws = (char*)d_ws;
  float* ht   = (float*)ws; ws += (size_t)BSZ * 128 * 4;
  float* rwT  = (float*)ws; ws += (size_t)NLAY * N_REL * BSZ * 4;
  float* hidA = (float*)ws; ws += (size_t)N_ENT * ROW * 4;
  float* hidB = (float*)ws; ws += (size_t)N_ENT * ROW * 4;
  float* emb  = (float*)ws; ws += (size_t)BSZ * 128 * 4;
  int*   counts = (int*)ws; ws += (size_t)N_ENT * 4;
  int*   rowst  = (int*)ws; ws += (size_t)(N_ENT + 1) * 4 + 12;
  int*   csr    = (int*)ws;

  // Stage 1: embeddings + per-layer relation attention (transposed layout)
  embed_kernel<<<dim3(8, 4), 32, 0, stream>>>(head, tail, ent_kg, Went_w, Went_b, ht);
  rw_kernel<<<NLAY * BSZ, 128, 0, stream>>>(ht, rl_w, rl_b, ar_w, ar_b, rwT);

  // Stage 2: CSR by destination (built fresh every call; deterministic)
  zero_int_kernel<<<(N_ENT + 255) / 256, 256, 0, stream>>>(counts, N_ENT);
  csr_count_kernel<<<(NEDGE + 255) / 256, 256, 0, stream>>>(edge_dst, counts, NEDGE);
  csr_scan_kernel<<<1, 32, 0, stream>>>(counts, rowst, N_ENT);
  csr_fill_kernel<<<(N_ENT + 127) / 128, 128, 0, stream>>>(edge_dst, rowst, csr, N_ENT, NEDGE);

  // Stage 3: two propagation directions, 3 layers each
  for (int dir = 0; dir < 2; ++dir) {
    const int* init_idx = (dir == 0) ? head : tail;
    const int* gath_idx = (dir == 0) ? tail : head;
    int init_off = (dir == 0) ? 0 : N_DIM;    // head_embed | tail_embed slice of ht
    int emb_off  = (dir == 0) ? N_DIM : 0;    // tail_hid -> [64:], head_hid -> [:64]

    zero_f4_kernel<<<(N_ENT * ROW) / (256 * 4), 256, 0, stream>>>((float4*)hidA);
    scatter_init_kernel<<<ROW / 256, 256, 0, stream>>>(hidA, ht, init_idx, init_off);

    for (int l = 0; l < NLAY; ++l) {
      aggregate_kernel<<<N_ENT, 256, 0, stream>>>(
          hidA, rel_kg + (size_t)l * N_REL * N_DIM, rwT + (size_t)l * N_REL * BSZ,
          edge_w, edge_src, edge_rel, rowst, csr, hidB);
      lin_gemm_kernel<<<(N_ENT * BSZ) / 16, 128, 0, stream>>>(
          hidB, lin_w + (size_t)l * N_DIM * N_DIM, lin_b + (size_t)l * N_DIM, hidA);
    }
    gather_kernel<<<ROW / 256, 256, 0, stream>>>(hidA, gath_idx, emb, emb_off);
  }

  // Stage 4: scoring head
  final_kernel<<<4, 256, 0, stream>>>(emb, Wr1_w, Wr1_b, out);
}